// RegionLoss_18176301597029
// MI455X (gfx1250) — compile-verified
//
#include <hip/hip_runtime.h>
#include <math.h>

// Problem constants (from reference)
#define NB    16
#define NA    5
#define NH    76
#define NW    76
#define MAXB  50
#define HW    (NH*NW)        // 5776
#define MCELLS (NA*HW)       // 28880 cells per batch
#define BLK2  256
#define NBX   113            // ceil(28880/256)
#define NPART (NBX*NB)       // 1808 block partials
#define NPAD  1856           // 29*64, padded for WMMA reduction

// Per-batch GT block layout (float offsets within one 640-float block).
// Contiguous per batch so one TDM descriptor can stage the whole block.
#define GT_XL   0            // gt box left edge        [50]
#define GT_XR   50           // gt box right edge       [50]
#define GT_YL   100          // gt box top edge         [50]
#define GT_YR   150          // gt box bottom edge      [50]
#define GT_AR   200          // gt box area             [50]
#define GT_VAL  250          // valid flag 0/1          [50]
#define GT_IDX  300          // assigned cell idx (int) [50], -1 if invalid
#define GT_TX   350
#define GT_TY   400
#define GT_TW   450
#define GT_TH   500
#define GT_TC   550
#define GTSTR   640          // padded block stride (2560 bytes)
#define WS_PART (NB*GTSTR)   // 10240: block partials [NPAD] follow
// total ws = 10240 + 1856 = 12096 floats = 48.4 KB

typedef __attribute__((ext_vector_type(2))) float        v2f;
typedef __attribute__((ext_vector_type(8))) float        v8f;
typedef __attribute__((ext_vector_type(4))) unsigned int v4u;
typedef __attribute__((ext_vector_type(4))) int          v4i;
typedef __attribute__((ext_vector_type(8))) int          v8i;

__device__ __forceinline__ float sigmoidf_(float v) { return 1.0f / (1.0f + expf(-v)); }

// ---------------------------------------------------------------------------
// Phase 1: per-GT preprocessing (16 blocks x 64 threads; thread t = GT slot)
// ---------------------------------------------------------------------------
__global__ __launch_bounds__(64)
void region_prep(const float* __restrict__ out, const float* __restrict__ tgt,
                 const float* __restrict__ anchors, float* __restrict__ ws)
{
    const int b = blockIdx.x;
    const int t = threadIdx.x;

    // zero the WMMA padding tail of the partial array (once, by block 0)
    if (b == 0 && t < (NPAD - NPART)) ws[WS_PART + NPART + t] = 0.0f;
    if (t >= MAXB) return;

    const float* g = tgt + (b * MAXB + t) * 5;
    const float x = g[1], y = g[2], wv = g[3], hv = g[4];

    // valid = cumprod(x != 0) up to and including t
    bool valid = true;
    for (int s = 0; s <= t; ++s)
        valid = valid && (tgt[(b * MAXB + s) * 5 + 1] != 0.0f);

    const float gx = x * (float)NW;
    const float gy = y * (float)NH;
    const float gw = valid ? wv * (float)NW : 1.0f;
    const float gh = valid ? hv * (float)NH : 1.0f;

    // best anchor (first max, strict >)
    int bestn = 0; float best = -1.0f;
    for (int a = 0; a < NA; ++a) {
        const float aw = anchors[2*a], ah = anchors[2*a+1];
        const float inter = fminf(gw, aw) * fminf(gh, ah);
        const float iou = inter / (gw*gh + aw*ah - inter);
        if (iou > best) { best = iou; bestn = a; }
    }

    const int gi = (int)gx;
    const int gj = (int)gy;
    const float aw = anchors[2*bestn], ah = anchors[2*bestn+1];

    const float txu = gx - (float)gi;
    const float tyu = gy - (float)gj;
    const float twu = logf(gw / aw);
    const float thu = logf(gh / ah);

    // decode predicted box at the assigned cell
    const int ob = (b * (NA*5) + bestn * 5) * HW + gj * NW + gi;
    const float px = sigmoidf_(out[ob])          + (float)gi;
    const float py = sigmoidf_(out[ob +   HW])   + (float)gj;
    const float pw = expf(out[ob + 2*HW]) * aw;
    const float ph = expf(out[ob + 3*HW]) * ah;

    // tconf = IoU(gt, pred-at-cell)
    const float iw = fminf(gx + gw*0.5f, px + pw*0.5f) - fmaxf(gx - gw*0.5f, px - pw*0.5f);
    const float ih = fminf(gy + gh*0.5f, py + ph*0.5f) - fmaxf(gy - gh*0.5f, py - ph*0.5f);
    const float inter = fmaxf(iw, 0.0f) * fmaxf(ih, 0.0f);
    const float tcu = inter / (gw*gh + pw*ph - inter);

    float* blk = ws + b * GTSTR;
    blk[GT_XL  + t] = gx - gw*0.5f;
    blk[GT_XR  + t] = gx + gw*0.5f;
    blk[GT_YL  + t] = gy - gh*0.5f;
    blk[GT_YR  + t] = gy + gh*0.5f;
    blk[GT_AR  + t] = gw * gh;
    blk[GT_VAL + t] = valid ? 1.0f : 0.0f;
    ((int*)blk)[GT_IDX + t] = valid ? (bestn * HW + gj * NW + gi) : -1;
    blk[GT_TX + t] = txu;
    blk[GT_TY + t] = tyu;
    blk[GT_TW + t] = twu;
    blk[GT_TH + t] = thu;
    blk[GT_TC + t] = tcu;
}

// ---------------------------------------------------------------------------
// Phase 2: per-cell decode + max-IoU vs 50 GT boxes + loss terms
// grid = (113, 16), block = 256 (8 wave32).
// GT block staged global->LDS with the Tensor Data Mover when available.
// ---------------------------------------------------------------------------
__global__ __launch_bounds__(BLK2)
void region_main(const float* __restrict__ out, const float* __restrict__ anchors,
                 const float* __restrict__ ws, float* __restrict__ part)
{
    __shared__ __align__(16) float sgt[GTSTR];   // staged per-batch GT block
    __shared__ float red[BLK2];

    const int b   = blockIdx.y;
    const int tid = threadIdx.x;
    const float* blk = ws + b * GTSTR;

#if __has_builtin(__builtin_amdgcn_tensor_load_to_lds)
    // ---- TDM staging: one 1x640 f32 tile, global -> LDS ----
    if (tid < 32) {
        const unsigned long long ga = (unsigned long long)(const void*)blk;
        const unsigned ldsOff = (unsigned)(unsigned long long)(const void*)&sgt[0];
        v4u g0;                              // D# group 0 (ISA 8.3)
        g0.x = 1u;                           // count=1 valid descriptor
        g0.y = ldsOff;                       // lds_addr [63:32]
        g0.z = (unsigned)ga;                 // global_addr [95:64]
        g0.w = ((unsigned)(ga >> 32) & 0x01FFFFFFu) | 0x80000000u; // addr hi | type=2
        v8i g1;                              // D# group 1 (ISA 8.4)
        g1[0] = 0x00020000;                  // workgroup_mask=0, data_size=2 (4B)
        g1[1] = (GTSTR & 0xFFFF) << 16;      // tensor_dim0 lo16 (=640)
        g1[2] = 1 << 16;                     // tensor_dim0 hi16=0 | tensor_dim1 lo16 (=1)
        g1[3] = GTSTR << 16;                 // tensor_dim1 hi16=0 | tile_dim0 (=640)
        g1[4] = 1;                           // tile_dim1=1 | tile_dim2=0
        g1[5] = GTSTR;                       // tensor_dim0_stride lo32 (=640)
        g1[6] = 0;                           // dim0_stride hi16 | dim1_stride lo16
        g1[7] = 0;                           // tensor_dim1_stride hi32
        v4i z4 = {0, 0, 0, 0};               // groups 2/3: unused (<=2D tensor)
        v8i z8 = {0, 0, 0, 0, 0, 0, 0, 0};   // extra group (6-arg toolchain form)
        __builtin_amdgcn_tensor_load_to_lds(g0, g1, z4, z4, z8, 0);
    }
    __builtin_amdgcn_s_wait_tensorcnt(0);    // waves that didn't issue: TENSORcnt==0
#else
    // ---- fallback staging: explicit LDS stores ----
    if (tid < GTSTR / 4) {
        ((float4*)sgt)[tid] = ((const float4*)blk)[tid];
    }
#endif
    __syncthreads();

    const float* sxl = sgt + GT_XL;
    const float* sxr = sgt + GT_XR;
    const float* syl = sgt + GT_YL;
    const float* syr = sgt + GT_YR;
    const float* sar = sgt + GT_AR;
    const float* sval = sgt + GT_VAL;
    const int*   sidx = (const int*)(sgt + GT_IDX);
    const float* stx = sgt + GT_TX;
    const float* sty = sgt + GT_TY;
    const float* stw = sgt + GT_TW;
    const float* sth = sgt + GT_TH;
    const float* stc = sgt + GT_TC;

    const int r = blockIdx.x * BLK2 + tid;
    float loss = 0.0f;

    if (r < MCELLS) {
        const int a = r / HW;
        const int p = r % HW;
        const int j = p / NW;
        const int i = p % NW;

        const int ob = (b * (NA*5) + a * 5) * HW + p;
        __builtin_prefetch(&out[ob + 4*HW], 0, 3);   // global_prefetch_b8 of conf plane

        const float xo = out[ob];
        const float yo = out[ob +   HW];
        const float wo = out[ob + 2*HW];
        const float ho = out[ob + 3*HW];
        const float co = out[ob + 4*HW];

        const float x    = sigmoidf_(xo);
        const float y    = sigmoidf_(yo);
        const float conf = sigmoidf_(co);

        const float aw = anchors[2*a], ah = anchors[2*a+1];
        const float px = x + (float)i;
        const float py = y + (float)j;
        const float pw = expf(wo) * aw;
        const float ph = expf(ho) * ah;

        const float pxl = px - pw*0.5f, pxr = px + pw*0.5f;
        const float pyl = py - ph*0.5f, pyr = py + ph*0.5f;
        const float parea = pw * ph;

        // max IoU against all GT boxes (invalid entries contribute 0)
        float maxiou = 0.0f;
        #pragma unroll 5
        for (int t = 0; t < MAXB; ++t) {
            const float iw = fminf(pxr, sxr[t]) - fmaxf(pxl, sxl[t]);
            const float ih = fminf(pyr, syr[t]) - fmaxf(pyl, syl[t]);
            const float inter = fmaxf(iw, 0.0f) * fmaxf(ih, 0.0f);
            float iou = inter / (parea + sar[t] - inter);
            iou *= sval[t];
            maxiou = fmaxf(maxiou, iou);
        }

        // last-match wins scan of the scatter list (emulates sequential .at[].set)
        int hit = -1;
        for (int t = 0; t < MAXB; ++t)
            if (sidx[t] == r) hit = t;

        float cfm2, tc;
        if (hit >= 0) {
            cfm2 = 5.0f;               // OBJECT_SCALE (conf_mask squared after sqrt)
            tc   = stc[hit];
            const float dx = x  - stx[hit];
            const float dy = y  - sty[hit];
            const float dw = wo - stw[hit];
            const float dh = ho - sth[hit];
            loss += dx*dx + dy*dy + dw*dw + dh*dh;   // coord_mask == 1 here
        } else {
            cfm2 = (maxiou > 0.6f) ? 0.0f : 1.0f;    // NOOBJECT_SCALE
            tc   = 0.0f;
        }
        const float dc = conf - tc;
        loss += cfm2 * dc * dc;
    }

    // deterministic block tree reduction
    red[tid] = loss;
    __syncthreads();
    for (int off = BLK2 >> 1; off > 0; off >>= 1) {
        if (tid < off) red[tid] += red[tid + off];
        __syncthreads();
    }
    if (tid == 0) part[blockIdx.y * NBX + blockIdx.x] = red[0];
}

// ---------------------------------------------------------------------------
// Phase 3: reduce 1808 (padded to 1856) partials via V_WMMA_F32_16X16X4_F32.
// One wave32. A = 64 partials per step, B = ones -> D[m][n] = sum_k A[m][k].
// Column N=0 lives in lanes 0 (M=0..7) and 16 (M=8..15) across the 8 C VGPRs,
// so total = lane0_sum + lane16_sum.
// ---------------------------------------------------------------------------
__global__ __launch_bounds__(32)
void region_final(const float* __restrict__ part, float* __restrict__ res)
{
    const int lane = threadIdx.x;
    v8f c = {};
    v2f ones; ones.x = 1.0f; ones.y = 1.0f;

    for (int base = 0; base < NPAD; base += 64) {
        v2f a;
        a.x = part[base + lane];
        a.y = part[base + 32 + lane];
        c = __builtin_amdgcn_wmma_f32_16x16x4_f32(
                /*neg_a=*/false, a, /*neg_b=*/false, ones,
                /*c_mod=*/(short)0, c, /*reuse_a=*/false, /*reuse_b=*/false);
    }

    const float s = c[0] + c[1] + c[2] + c[3] + c[4] + c[5] + c[6] + c[7];
    const float t0  = __shfl(s, 0, 32);
    const float t16 = __shfl(s, 16, 32);
    if (lane == 0) res[0] = 0.5f * (t0 + t16);
}

// ---------------------------------------------------------------------------
extern "C" void kernel_launch(void* const* d_in, const int* in_sizes, int n_in,
                              void* d_out, int out_size, void* d_ws, size_t ws_size,
                              hipStream_t stream) {
    const float* out     = (const float*)d_in[0];   // (16, 25, 76, 76) f32
    const float* tgt     = (const float*)d_in[1];   // (16, 250) f32
    const float* anchors = (const float*)d_in[2];   // (10,) f32
    float* ws  = (float*)d_ws;
    float* res = (float*)d_out;

    region_prep <<<NB, 64, 0, stream>>>(out, tgt, anchors, ws);
    region_main <<<dim3(NBX, NB), BLK2, 0, stream>>>(out, anchors, ws, ws + WS_PART);
    region_final<<<1, 32, 0, stream>>>(ws + WS_PART, res);
}